// Soft_Collision_Loss_23167053595265
// MI455X (gfx1250) — compile-verified
//
#include <hip/hip_runtime.h>

#define NVERT  6400
#define NFACE  12482
#define NGARM  8192
#define NBATCH 2
#define TRI_SPLIT 8
#define TRI_PER_SPLIT ((NFACE + TRI_SPLIT - 1) / TRI_SPLIT)
#define TILE_TRIS 64

typedef __attribute__((ext_vector_type(2))) float v2f;
typedef __attribute__((ext_vector_type(4))) float v4f;
typedef __attribute__((ext_vector_type(8))) float v8f;

struct F3 { float x, y, z; };
__device__ __forceinline__ F3 sub3(F3 a, F3 b){ return {a.x-b.x, a.y-b.y, a.z-b.z}; }
__device__ __forceinline__ F3 add3(F3 a, F3 b){ return {a.x+b.x, a.y+b.y, a.z+b.z}; }
__device__ __forceinline__ F3 mul3(F3 a, float s){ return {a.x*s, a.y*s, a.z*s}; }
__device__ __forceinline__ float dot3(F3 a, F3 b){ return a.x*b.x + a.y*b.y + a.z*b.z; }
__device__ __forceinline__ F3 cross3(F3 a, F3 b){
  return {a.y*b.z - a.z*b.y, a.z*b.x - a.x*b.z, a.x*b.y - a.y*b.x};
}
__device__ __forceinline__ float safef(float x){ return (fabsf(x) > 1e-20f) ? x : 1.0f; }

__device__ __forceinline__ v8f wmma_f32x4(v2f a, v2f b, v8f c){
  // D(16x16,f32) = A(16x4,f32) * B(4x16,f32) + C
  return __builtin_amdgcn_wmma_f32_16x16x4_f32(false, a, false, b, (short)0, c,
                                               false, false);
}

// ---------------------------------------------------------------- init
__global__ void k_init(unsigned long long* __restrict__ best,
                       float* __restrict__ vn, float* __restrict__ out)
{
  int i = blockIdx.x * blockDim.x + threadIdx.x;
  if (i < NBATCH * NGARM)     best[i] = 0xFFFFFFFFFFFFFFFFull;
  if (i < NBATCH * NVERT * 3) vn[i]   = 0.0f;
  if (i < NBATCH)             out[i]  = 0.0f;
}

// --------------------------------------------------- per-triangle precompute
// tri pack (16 floats): [0..3]  (ab.x, ab.y, ab.z, -ab.a)   -> s row
//                       [4..7]  (ac.x, ac.y, ac.z, -ac.a)   -> t row
//                       [8..11] (-2a.x,-2a.y,-2a.z,  a.a)   -> u row
//                       [12..14] aa, bb, abac   [15] pad
__global__ void k_tripre(const float* __restrict__ bv, const int* __restrict__ faces,
                         float* __restrict__ fn, float* __restrict__ vn,
                         float* __restrict__ tp)
{
  int tid = blockIdx.x * blockDim.x + threadIdx.x;
  if (tid >= NBATCH * NFACE) return;
  int b = tid / NFACE, f = tid % NFACE;
  int i0 = faces[f*3+0], i1 = faces[f*3+1], i2 = faces[f*3+2];
  const float* bvb = bv + (size_t)b * NVERT * 3;
  F3 a  = {bvb[i0*3], bvb[i0*3+1], bvb[i0*3+2]};
  F3 bb = {bvb[i1*3], bvb[i1*3+1], bvb[i1*3+2]};
  F3 c  = {bvb[i2*3], bvb[i2*3+1], bvb[i2*3+2]};
  F3 ab = sub3(bb, a), ac = sub3(c, a);
  F3 fx = cross3(ab, ac);
  float ln = sqrtf(dot3(fx, fx)) + 1e-10f;
  float* fno = fn + (size_t)tid * 3;
  fno[0] = fx.x / ln; fno[1] = fx.y / ln; fno[2] = fx.z / ln;
  float* vnb = vn + (size_t)b * NVERT * 3;
  atomicAdd(&vnb[i0*3+0], fx.x); atomicAdd(&vnb[i0*3+1], fx.y); atomicAdd(&vnb[i0*3+2], fx.z);
  atomicAdd(&vnb[i1*3+0], fx.x); atomicAdd(&vnb[i1*3+1], fx.y); atomicAdd(&vnb[i1*3+2], fx.z);
  atomicAdd(&vnb[i2*3+0], fx.x); atomicAdd(&vnb[i2*3+1], fx.y); atomicAdd(&vnb[i2*3+2], fx.z);
  float* t = tp + (size_t)tid * 16;
  t[0] = ab.x;      t[1] = ab.y;      t[2] = ab.z;      t[3]  = -dot3(ab, a);
  t[4] = ac.x;      t[5] = ac.y;      t[6] = ac.z;      t[7]  = -dot3(ac, a);
  t[8] = -2.f*a.x;  t[9] = -2.f*a.y;  t[10] = -2.f*a.z; t[11] =  dot3(a, a);
  t[12] = dot3(ab, ab); t[13] = dot3(ac, ac); t[14] = dot3(ab, ac); t[15] = 0.f;
}

__global__ void k_vnorm(float* __restrict__ vn)
{
  int i = blockIdx.x * blockDim.x + threadIdx.x;
  if (i >= NBATCH * NVERT) return;
  float x = vn[i*3], y = vn[i*3+1], z = vn[i*3+2];
  float ln = sqrtf(x*x + y*y + z*z) + 1e-10f;
  vn[i*3] = x/ln; vn[i*3+1] = y/ln; vn[i*3+2] = z/ln;
}

// --------------------------------------------------- WMMA brute-force argmin
// grid (NGARM/128, TRI_SPLIT, NBATCH), 256 threads = 8 wave32, 16 points/wave.
__global__ void __launch_bounds__(256)
k_argmin(const float* __restrict__ g, const float* __restrict__ tp,
         unsigned long long* __restrict__ best)
{
  const int wave = threadIdx.x >> 5;
  const int lane = threadIdx.x & 31;
  const int col  = lane & 15;            // B/D column = triangle within sub-tile
  const bool hi  = lane >= 16;
  const int b    = blockIdx.z;
  const int ptBase   = blockIdx.x * 128 + wave * 16;
  const int triBegin = blockIdx.y * TRI_PER_SPLIT;
  const int triEnd   = min(triBegin + TRI_PER_SPLIT, NFACE);

  __shared__ float ldsTile[TILE_TRIS * 16];   // 64 tris x 16 floats = 4KB
  __shared__ float ldsPP[8 * 16];             // |p|^2 per wave-point

  // A operand: lanes 0-15 hold (x,y) of point M=lane; lanes 16-31 hold (z,1).
  const int pid = ptBase + col;
  const float* gp = g + ((size_t)b * NGARM + pid) * 3;
  const float px = gp[0], py = gp[1], pz = gp[2];
  if (!hi) ldsPP[wave * 16 + col] = px*px + py*py + pz*pz;
  v2f A; A[0] = hi ? pz : px; A[1] = hi ? 1.0f : py;

  float dmin[8]; unsigned int didx[8];
#pragma unroll
  for (int m = 0; m < 8; ++m) { dmin[m] = 3.0e38f; didx[m] = 0u; }

  const float* tpB = tp + (size_t)b * NFACE * 16;

  for (int t0 = triBegin; t0 < triEnd; t0 += TILE_TRIS) {
    __syncthreads();
    { // cooperative tile load: 1024 floats, one float4 per thread
      int e = threadIdx.x;                  // 0..255
      int tri = t0 + (e >> 2);
      v4f val = {};
      if (tri < triEnd)
        val = *(const v4f*)(tpB + (size_t)tri * 16 + (e & 3) * 4);
      *(v4f*)(&ldsTile[e * 4]) = val;
    }
    { // prefetch next tile into cache (global_prefetch_b8)
      int nxt = t0 + TILE_TRIS;
      if (nxt < triEnd)
        __builtin_prefetch(tpB + (size_t)nxt * 16 + threadIdx.x * 4, 0, 3);
    }
    __syncthreads();

#pragma unroll
    for (int h = 0; h < TILE_TRIS / 16; ++h) {
      const float* tc = &ldsTile[(h * 16 + col) * 16];
      const v2f* tcv = (const v2f*)tc + (hi ? 1 : 0);
      v2f Bs = tcv[0];                      // ds_load_b64 offset 0 / 8
      v2f Bt = tcv[2];                      // offset 16 / 24
      v2f Bu = tcv[4];                      // offset 32 / 40
      const float aa = tc[12], bb = tc[13], abac = tc[14];

      v8f z = {};
      v8f S = wmma_f32x4(A, Bs, z);   // s = ab.(p-a)   for all 16x16 pairs
      v8f T = wmma_f32x4(A, Bt, z);   // t = ac.(p-a)
      v8f U = wmma_f32x4(A, Bu, z);   // u = -2 a.p + a.a

      const unsigned int triIdx = (unsigned int)(t0 + h * 16 + col);
      const bool valid = (int)triIdx < triEnd;

#pragma unroll
      for (int m = 0; m < 8; ++m) {
        float s = S[m], t = T[m], u = U[m];
        float d1 = s, d2 = t;
        float d3 = s - aa, d4 = t - abac, d5 = s - abac, d6 = t - bb;
        float va = fmaf(d3, d6, -d5 * d4);
        float vb = fmaf(d5, d2, -d1 * d6);
        float vc = fmaf(d1, d4, -d3 * d2);
        float e43 = d4 - d3, e56 = d5 - d6;
        // region chain in reference order (last matching condition wins),
        // fully branchless: each condition folds into per-variable selects
        bool cBC = (va <= 0.f) & (e43 >= 0.f) & (e56 >= 0.f);
        bool cCA = (vb <= 0.f) & (d2 >= 0.f) & (d6 <= 0.f);
        bool cC  = (d6 >= 0.f) & (d5 <= d6);
        bool cAB = (vc <= 0.f) & (d1 >= 0.f) & (d3 <= 0.f);
        bool cB  = (d3 >= 0.f) & (d4 <= d3);
        bool cA  = (d1 <= 0.f) & (d2 <= 0.f);
        float nv  = vb,  nw  = vc,  den = va + vb + vc;        // interior
        nv = cBC ? e56 : nv;  nw = cBC ? e43 : nw;  den = cBC ? (e43 + e56) : den;
        nv = cCA ? 0.f : nv;  nw = cCA ? d2  : nw;  den = cCA ? (d2 - d6)   : den;
        nv = cC  ? 0.f : nv;  nw = cC  ? 1.f : nw;  den = cC  ? 1.f         : den;
        nv = cAB ? d1  : nv;  nw = cAB ? 0.f : nw;  den = cAB ? (d1 - d3)   : den;
        nv = cB  ? 1.f : nv;  nw = cB  ? 0.f : nw;  den = cB  ? 1.f         : den;
        nv = cA  ? 0.f : nv;  nw = cA  ? 0.f : nw;  den = cA  ? 1.f         : den;
        den = safef(den);
        float r = __builtin_amdgcn_rcpf(den);
        float v = nv * r, w = nw * r;
        // dist^2 - |p|^2  =  u + v^2 aa + 2vw abac - 2vs + w^2 bb - 2wt
        float dd = u + v * (fmaf(v, aa, 2.0f * w * abac) - 2.0f * s)
                     + w * fmaf(w, bb, -2.0f * t);
        bool upd = valid & (dd < dmin[m]);
        dmin[m] = upd ? dd : dmin[m];
        didx[m] = upd ? triIdx : didx[m];
      }
    }
  }
  __syncthreads();

#pragma unroll
  for (int m = 0; m < 8; ++m) {
    int mi = hi ? (8 + m) : m;            // D row -> point within wave tile
    int p  = ptBase + mi;
    float pp = ldsPP[wave * 16 + mi];
    float dist = fmaxf(dmin[m] + pp, 0.0f);
    unsigned long long key =
        ((unsigned long long)__float_as_uint(dist) << 32) | (unsigned long long)didx[m];
    atomicMin(&best[(size_t)b * NGARM + p], key);
  }
}

// --------------------------------------------------- exact recompute + loss
__device__ void closest_pt(F3 p, F3 a, F3 b, F3 c, F3& cp, int& reg)
{
  F3 ab = sub3(b, a), ac = sub3(c, a), ap = sub3(p, a);
  float d1 = dot3(ab, ap), d2 = dot3(ac, ap);
  F3 bp = sub3(p, b); float d3 = dot3(ab, bp), d4 = dot3(ac, bp);
  F3 cp2 = sub3(p, c); float d5 = dot3(ab, cp2), d6 = dot3(ac, cp2);
  float va = d3*d6 - d5*d4, vb = d5*d2 - d1*d6, vc = d1*d4 - d3*d2;
  float denom = safef(va + vb + vc);
  float v = vb / denom, w = vc / denom;
  cp = add3(a, add3(mul3(ab, v), mul3(ac, w))); reg = 0;
  float t;
  if (va <= 0.f && (d4 - d3) >= 0.f && (d5 - d6) >= 0.f) {
    t = (d4 - d3) / safef((d4 - d3) + (d5 - d6));
    cp = add3(b, mul3(sub3(c, b), t)); reg = 2;
  }
  if (vb <= 0.f && d2 >= 0.f && d6 <= 0.f) {
    t = d2 / safef(d2 - d6); cp = add3(a, mul3(ac, t)); reg = 3;
  }
  if (d6 >= 0.f && d5 <= d6) { cp = c; reg = 6; }
  if (vc <= 0.f && d1 >= 0.f && d3 <= 0.f) {
    t = d1 / safef(d1 - d3); cp = add3(a, mul3(ab, t)); reg = 1;
  }
  if (d3 >= 0.f && d4 <= d3) { cp = b; reg = 5; }
  if (d1 <= 0.f && d2 <= 0.f) { cp = a; reg = 4; }
}

__global__ void __launch_bounds__(256)
k_final(const float* __restrict__ g, const float* __restrict__ bv,
        const int* __restrict__ faces, const unsigned long long* __restrict__ best,
        const float* __restrict__ fn, const float* __restrict__ vn,
        float* __restrict__ out)
{
  int b = blockIdx.y;
  int i = blockIdx.x * blockDim.x + threadIdx.x;   // 0..NGARM-1
  const float* gp = g + ((size_t)b * NGARM + i) * 3;
  F3 p = {gp[0], gp[1], gp[2]};
  unsigned int idx = (unsigned int)(best[(size_t)b * NGARM + i] & 0xFFFFFFFFu);
  if (idx >= NFACE) idx = 0;
  int i0 = faces[idx*3+0], i1 = faces[idx*3+1], i2 = faces[idx*3+2];
  const float* bvb = bv + (size_t)b * NVERT * 3;
  F3 a  = {bvb[i0*3], bvb[i0*3+1], bvb[i0*3+2]};
  F3 bb = {bvb[i1*3], bvb[i1*3+1], bvb[i1*3+2]};
  F3 c  = {bvb[i2*3], bvb[i2*3+1], bvb[i2*3+2]};
  F3 cp; int reg;
  closest_pt(p, a, bb, c, cp, reg);
  const float* vnb = vn + (size_t)b * NVERT * 3;
  F3 n;
  if (reg == 0) {
    const float* f0 = fn + ((size_t)b * NFACE + idx) * 3;
    n = {f0[0], f0[1], f0[2]};
  } else if (reg >= 4) {
    int j = reg - 4; j = j < 0 ? 0 : (j > 2 ? 2 : j);
    int vI = (j == 0) ? i0 : ((j == 1) ? i1 : i2);
    n = {vnb[vI*3], vnb[vI*3+1], vnb[vI*3+2]};
  } else {
    int e = reg - 1; e = e < 0 ? 0 : (e > 2 ? 2 : e);
    int e2 = (e + 1) % 3;
    int vA = (e  == 0) ? i0 : ((e  == 1) ? i1 : i2);
    int vB = (e2 == 0) ? i0 : ((e2 == 1) ? i1 : i2);
    n = {vnb[vA*3] + vnb[vB*3], vnb[vA*3+1] + vnb[vB*3+1], vnb[vA*3+2] + vnb[vB*3+2]};
  }
  float ln = sqrtf(dot3(n, n)) + 1e-10f;
  n = {n.x / ln, n.y / ln, n.z / ln};
  float d = fmaxf(dot3(sub3(cp, p), n), 0.0f);   // relu(-(g-cp).n)

  __shared__ float red[256];
  red[threadIdx.x] = d;
  __syncthreads();
  for (int s = 128; s > 0; s >>= 1) {
    if (threadIdx.x < s) red[threadIdx.x] += red[threadIdx.x + s];
    __syncthreads();
  }
  if (threadIdx.x == 0) atomicAdd(&out[b], red[0]);
}

// ---------------------------------------------------------------- launch
extern "C" void kernel_launch(void* const* d_in, const int* in_sizes, int n_in,
                              void* d_out, int out_size, void* d_ws, size_t ws_size,
                              hipStream_t stream)
{
  (void)in_sizes; (void)n_in; (void)out_size; (void)ws_size;
  const float* gv    = (const float*)d_in[0];   // (2, 8192, 3) f32
  const float* bv    = (const float*)d_in[1];   // (2, 6400, 3) f32
  const int*   faces = (const int*)d_in[2];     // (12482, 3) i32
  float* out = (float*)d_out;                   // (2,) f32

  char* ws = (char*)d_ws;
  size_t off = 0;
  auto carve = [&](size_t bytes) {
    size_t o = off;
    off = (off + bytes + 255) & ~(size_t)255;
    return o;
  };
  unsigned long long* best = (unsigned long long*)(ws + carve((size_t)NBATCH * NGARM * 8));
  float* vn = (float*)(ws + carve((size_t)NBATCH * NVERT * 3 * 4));
  float* fn = (float*)(ws + carve((size_t)NBATCH * NFACE * 3 * 4));
  float* tp = (float*)(ws + carve((size_t)NBATCH * NFACE * 16 * 4));

  {
    int n = NBATCH * NVERT * 3;     // largest init span
    k_init<<<(n + 255) / 256, 256, 0, stream>>>(best, vn, out);
  }
  {
    int n = NBATCH * NFACE;
    k_tripre<<<(n + 255) / 256, 256, 0, stream>>>(bv, faces, fn, vn, tp);
  }
  {
    int n = NBATCH * NVERT;
    k_vnorm<<<(n + 255) / 256, 256, 0, stream>>>(vn);
  }
  {
    dim3 grid(NGARM / 128, TRI_SPLIT, NBATCH);
    k_argmin<<<grid, 256, 0, stream>>>(gv, tp, best);
  }
  {
    dim3 grid(NGARM / 256, NBATCH);
    k_final<<<grid, 256, 0, stream>>>(gv, bv, faces, best, fn, vn, out);
  }
}